// TV_41832981463242
// MI455X (gfx1250) — compile-verified
//
#include <hip/hip_runtime.h>
#include <cstdint>

// ---------------------------------------------------------------------------
// Chambolle TV dual iteration, 4096x4096 f32, 50 iterations + final pass.
// Memory-bound: ~16 GB total traffic -> ~0.7 ms at 23.3 TB/s HBM.
// Uses CDNA5 async global->LDS loads (ASYNCcnt path) for halo-tile staging,
// b128-wide for the aligned x / p0 tiles, b32 for the misaligned p1 tile.
// ---------------------------------------------------------------------------

#define TV_H 4096
#define TV_W 4096
#define TV_TS 32          // output tile (32x32)
#define TV_BTH 256        // 8 wave32 per block

// LDS row strides (floats). 36 -> 144 B rows, 16 B aligned for b128 stores.
#define XS_S 36
#define P0_S 36
#define P1_S 35

static constexpr float TV_C   = 0.1f;   // ALPHA * LAMDA
static constexpr float TV_TAU = 0.25f;
static constexpr float TV_TOL = 1e-4f;

// ---- CDNA5 async global->LDS staging (gfx1250 device pass only) -----------
// Hard requirement on the gfx1250 pass: a successful compile proves the
// CDNA5 async-to-LDS instructions were emitted (no silent scalar fallback).
#if defined(__gfx1250__)
#if !__has_builtin(__builtin_amdgcn_global_load_async_to_lds_b32) || !__has_builtin(__builtin_amdgcn_s_wait_asynccnt)
#error "gfx1250 async-to-LDS builtins unavailable on this toolchain"
#endif
#define TV_ASYNC_LDS 1
#if __has_builtin(__builtin_amdgcn_global_load_async_to_lds_b128)
#define TV_ASYNC_B128 1
#else
#define TV_ASYNC_B128 0
#endif
#else
#define TV_ASYNC_LDS 0
#define TV_ASYNC_B128 0
#endif

#if TV_ASYNC_LDS
typedef int tv_v4i __attribute__((ext_vector_type(4)));
typedef __attribute__((address_space(1))) int     GI;
typedef __attribute__((address_space(3))) int     LI;
typedef __attribute__((address_space(1))) tv_v4i  GV4;
typedef __attribute__((address_space(3))) tv_v4i  LV4;
#endif

__device__ __forceinline__ void tv_g2l_b32(const float* gsrc, float* ldst) {
#if TV_ASYNC_LDS
  // per-lane gather: copies 4 bytes from the global address in gsrc to the LDS address in ldst
  __builtin_amdgcn_global_load_async_to_lds_b32(
      (GI*)(uintptr_t)gsrc,
      (LI*)(uintptr_t)ldst,   // low 32 bits of generic addr = LDS offset
      0, 0);
#else
  *ldst = *gsrc;
#endif
}

__device__ __forceinline__ void tv_g2l_b128(const float* gsrc, float* ldst) {
#if TV_ASYNC_B128
  __builtin_amdgcn_global_load_async_to_lds_b128(
      (GV4*)(uintptr_t)gsrc,
      (LV4*)(uintptr_t)ldst, 0, 0);
#else
  tv_g2l_b32(gsrc + 0, ldst + 0);
  tv_g2l_b32(gsrc + 1, ldst + 1);
  tv_g2l_b32(gsrc + 2, ldst + 2);
  tv_g2l_b32(gsrc + 3, ldst + 3);
#endif
}

__device__ __forceinline__ void tv_async_wait() {
#if TV_ASYNC_LDS
  __builtin_amdgcn_s_wait_asynccnt(0);
#endif
}

// Stage a (rows x 33) row-aligned tile (x or p0): global rows are 128 B
// aligned (tile col base is a multiple of 32), LDS rows are 16 B aligned.
// Per row: 8x b128 (cols 0..31) + 1x b32 (halo col 32, clamped at edge).
template <int ROWS>
__device__ __forceinline__ void tv_stage_aligned(const float* __restrict__ g,
                                                 float* __restrict__ lds,
                                                 int ldsStride, int r0row0,
                                                 int c0, int tid) {
  for (int idx = tid; idx < ROWS * 9; idx += TV_BTH) {
    const int ty = idx / 9, chunk = idx % 9;
    const int gi = min(max(r0row0 + ty, 0), TV_H - 1);
    if (chunk < 8) {
      tv_g2l_b128(&g[(size_t)gi * TV_W + c0 + 4 * chunk], &lds[ty * ldsStride + 4 * chunk]);
    } else {
      const int gj = min(c0 + 32, TV_W - 1);
      tv_g2l_b32(&g[(size_t)gi * TV_W + gj], &lds[ty * ldsStride + 32]);
    }
  }
}

// ---------------------------------------------------------------------------
// One dual iteration, fused:  u = x - C*div(p);  p' = clip(p + TAU*grad(u)).
// Clamp-to-edge staging reproduces the leading-boundary zero pads of div();
// the trailing-boundary zero pads of grad() are handled by explicit guards.
// ---------------------------------------------------------------------------
template <bool FIRST>
__global__ __launch_bounds__(TV_BTH)
void tv_iter(const float* __restrict__ x,
             const float* __restrict__ p0i, const float* __restrict__ p1i,
             float* __restrict__ p0o, float* __restrict__ p1o,
             unsigned* __restrict__ maxbits, const unsigned* __restrict__ donep) {
  __shared__ __align__(16) float p0s[34 * P0_S];  // rows r0-1..r0+32, cols c0..c0+32
  __shared__ __align__(16) float p1s[33 * P1_S];  // rows r0..r0+32, cols c0-1..c0+32
  __shared__ __align__(16) float xs [33 * XS_S];  // rows r0..r0+32, cols c0..c0+32
  __shared__ __align__(16) float us [33 * XS_S];

  const int tid = threadIdx.x;
  const int r0 = blockIdx.y * TV_TS;
  const int c0 = blockIdx.x * TV_TS;

  if (!FIRST) {
    // reference: p_next = done ? p : p_new  (done is from the PREVIOUS iter)
    const bool frozen = (*donep != 0u);
    if (frozen) {
      for (int idx = tid; idx < TV_TS * TV_TS; idx += TV_BTH) {
        const int ty = idx >> 5, tx = idx & 31;
        const size_t g = (size_t)(r0 + ty) * TV_W + (c0 + tx);
        p0o[g] = p0i[g];
        p1o[g] = p1i[g];
      }
      return;
    }
  }

  // ---- stage tiles into LDS (async on gfx1250, tracked by ASYNCcnt) ----
  tv_stage_aligned<33>(x, xs, XS_S, r0, c0, tid);
  if (!FIRST) {
    tv_stage_aligned<34>(p0i, p0s, P0_S, r0 - 1, c0, tid);   // +1 top halo row
    // p1 tile starts at col c0-1: misaligned -> b32 per element
    for (int idx = tid; idx < 33 * 34; idx += TV_BTH) {
      const int ty = idx / 34, tx = idx % 34;
      const int gi = min(r0 + ty, TV_H - 1);
      const int gj = min(max(c0 - 1 + tx, 0), TV_W - 1);
      tv_g2l_b32(&p1i[(size_t)gi * TV_W + gj], &p1s[ty * P1_S + tx]);
    }
  }
  tv_async_wait();
  __syncthreads();

  // ---- u = x - C*div(p) over the (TS+1)x(TS+1) halo region ----
  if (!FIRST) {
    for (int idx = tid; idx < 33 * 33; idx += TV_BTH) {
      const int ty = idx / 33, tx = idx % 33;
      // div: dx[i] = (i==0) ? 0 : p0[i-1]-p0[i]  -> duplicated clamp row gives 0
      const float dx = p0s[ty * P0_S + tx] - p0s[(ty + 1) * P0_S + tx];
      const float dy = p1s[ty * P1_S + tx] - p1s[ty * P1_S + tx + 1];
      us[ty * XS_S + tx] = xs[ty * XS_S + tx] - TV_C * (dx + dy);
    }
    __syncthreads();
  }
  const float* u = FIRST ? xs : us;   // p==0 on the first iteration -> u == x

  // ---- p update + per-wave max|dp| reduction ----
  float lmax = 0.f;
  for (int idx = tid; idx < TV_TS * TV_TS; idx += TV_BTH) {
    const int ty = idx >> 5, tx = idx & 31;
    const int gi = r0 + ty, gj = c0 + tx;
    const float uc = u[ty * XS_S + tx];
    const float gx = (gi < TV_H - 1) ? (u[(ty + 1) * XS_S + tx] - uc) : 0.f;
    const float gy = (gj < TV_W - 1) ? (u[ty * XS_S + tx + 1] - uc) : 0.f;
    const float p0c = FIRST ? 0.f : p0s[(ty + 1) * P0_S + tx];  // p0[gi][gj]
    const float p1c = FIRST ? 0.f : p1s[ty * P1_S + tx + 1];    // p1[gi][gj]
    const float p0n = fminf(fmaxf(p0c + TV_TAU * gx, -1.f), 1.f);
    const float p1n = fminf(fmaxf(p1c + TV_TAU * gy, -1.f), 1.f);
    lmax = fmaxf(lmax, fmaxf(fabsf(p0n - p0c), fabsf(p1n - p1c)));
    const size_t g = (size_t)gi * TV_W + gj;
    p0o[g] = p0n;
    p1o[g] = p1n;
  }

  // wave32 reduction; float bits are order-preserving for non-negative floats
  unsigned m = __float_as_uint(lmax);
  #pragma unroll
  for (int off = 16; off > 0; off >>= 1) {
    const unsigned o = (unsigned)__shfl_xor((int)m, off, 32);
    m = (o > m) ? o : m;
  }
  if ((tid & 31) == 0) atomicMax(maxbits, m);
}

// Fold this iteration's dp_max into the sticky done flag, reset accumulator.
__global__ void tv_check(unsigned* __restrict__ flags) {
  if (threadIdx.x == 0) {
    if (__uint_as_float(flags[0]) < TV_TOL) flags[1] = 1u;
    flags[0] = 0u;
  }
}

// out = x - C * div(p_final)
__global__ __launch_bounds__(TV_BTH)
void tv_final(const float* __restrict__ x,
              const float* __restrict__ p0, const float* __restrict__ p1,
              float* __restrict__ out) {
  const size_t idx = (size_t)blockIdx.x * TV_BTH + threadIdx.x;
  const int i = (int)(idx >> 12);
  const int j = (int)(idx & (TV_W - 1));
  const float dx = (i > 0) ? (p0[idx - TV_W] - p0[idx]) : 0.f;
  const float dy = (j > 0) ? (p1[idx - 1] - p1[idx]) : 0.f;
  out[idx] = x[idx] - TV_C * (dx + dy);
}

extern "C" void kernel_launch(void* const* d_in, const int* in_sizes, int n_in,
                              void* d_out, int out_size, void* d_ws, size_t ws_size,
                              hipStream_t stream) {
  (void)in_sizes; (void)n_in; (void)out_size; (void)ws_size;
  const float* x = (const float*)d_in[0];
  float* out = (float*)d_out;

  const size_t N = (size_t)TV_H * TV_W;
  float* p0A = (float*)d_ws;       // ping
  float* p1A = p0A + N;
  float* p0B = p1A + N;            // pong
  float* p1B = p0B + N;
  unsigned* flags = (unsigned*)(p1B + N);   // [0]=max|dp| bits, [1]=done

  (void)hipMemsetAsync(flags, 0, 2 * sizeof(unsigned), stream);  // capture-legal

  const dim3 grid(TV_W / TV_TS, TV_H / TV_TS);
  const dim3 block(TV_BTH);

  // iteration 0: p == 0 (p-input args are unread)
  tv_iter<true><<<grid, block, 0, stream>>>(x, p0A, p1A, p0A, p1A, flags, flags + 1);
  tv_check<<<1, 32, 0, stream>>>(flags);

  for (int k = 1; k < 50; ++k) {
    const bool inA = (k & 1);
    const float* pi0 = inA ? p0A : p0B;
    const float* pi1 = inA ? p1A : p1B;
    float* po0 = inA ? p0B : p0A;
    float* po1 = inA ? p1B : p1A;
    tv_iter<false><<<grid, block, 0, stream>>>(x, pi0, pi1, po0, po1, flags, flags + 1);
    if (k < 49) tv_check<<<1, 32, 0, stream>>>(flags);  // last dp_max is unused
  }

  // after iter 49 (in=A, out=B) the final p lives in buffer B
  tv_final<<<(unsigned)(N / TV_BTH), block, 0, stream>>>(x, p0B, p1B, out);
}